// ContrastiveSpeakerLoss_57518202028104
// MI455X (gfx1250) — compile-verified
//
#include <hip/hip_runtime.h>
#include <hip/hip_bf16.h>

// ContrastiveSpeakerLoss on MI455X (gfx1250, wave32, WMMA).
//
// B=4096, D=256, K=6 negatives, 512 speakers, margin 0.1.
// sim GEMM = 8.6 GFLOP f32; all data L2-resident (4 MB inputs << 192 MB L2),
// so we keep full f32 precision and use V_WMMA_F32_16X16X4_F32.

#define B_SZ   4096
#define D_SZ   256
#define K_NEG  6
#define MARGIN 0.1f

#define CHUNK  64     // K-chunk staged in LDS per iteration
#define LDP    68     // padded LDS row stride (floats): 68 % 64 = 4 -> conflict-free

typedef __attribute__((ext_vector_type(2))) float v2f;
typedef __attribute__((ext_vector_type(8))) float v8f;

// ---------------------------------------------------------------------------
// Kernel 1: row L2-normalize (one wave32 per row) + zero the accumulators.
// ---------------------------------------------------------------------------
__global__ __launch_bounds__(256) void csl_normalize(const float* __restrict__ x,
                                                     float* __restrict__ e,
                                                     float* __restrict__ accum) {
  if (blockIdx.x == 0 && threadIdx.x < 2) accum[threadIdx.x] = 0.0f;

  const int wave = threadIdx.x >> 5;
  const int lane = threadIdx.x & 31;
  const int row  = blockIdx.x * 8 + wave;          // 512 blocks * 8 waves = 4096

  const float4* xr = (const float4*)(x + (size_t)row * D_SZ);
  float4 a = xr[lane * 2 + 0];
  float4 b = xr[lane * 2 + 1];

  float ss = a.x * a.x + a.y * a.y + a.z * a.z + a.w * a.w
           + b.x * b.x + b.y * b.y + b.z * b.z + b.w * b.w;
  #pragma unroll
  for (int off = 16; off >= 1; off >>= 1) ss += __shfl_xor(ss, off, 32);

  const float nrm = fmaxf(sqrtf(ss), 1e-12f);      // matches jnp.maximum(norm, 1e-12)
  const float s   = 1.0f / nrm;

  a.x *= s; a.y *= s; a.z *= s; a.w *= s;
  b.x *= s; b.y *= s; b.z *= s; b.w *= s;
  float4* er = (float4*)(e + (size_t)row * D_SZ);
  er[lane * 2 + 0] = a;
  er[lane * 2 + 1] = b;
}

// ---------------------------------------------------------------------------
// Kernel 2: neg_sim[i,k] = dot(e[i], e[negidx[i,k]])  (one wave32 per (i,k)).
// ---------------------------------------------------------------------------
__global__ __launch_bounds__(256) void csl_negsim(const float* __restrict__ e,
                                                  const int* __restrict__ negidx,
                                                  float* __restrict__ negsim) {
  const int wave = threadIdx.x >> 5;
  const int lane = threadIdx.x & 31;
  const int g    = blockIdx.x * 8 + wave;          // 0 .. B*K-1
  const int i    = g / K_NEG;
  const int j    = negidx[g];

  const float4* xi = (const float4*)(e + (size_t)i * D_SZ);
  const float4* xj = (const float4*)(e + (size_t)j * D_SZ);
  float4 a0 = xi[lane * 2 + 0], b0 = xj[lane * 2 + 0];
  float4 a1 = xi[lane * 2 + 1], b1 = xj[lane * 2 + 1];

  float acc = a0.x * b0.x + a0.y * b0.y + a0.z * b0.z + a0.w * b0.w
            + a1.x * b1.x + a1.y * b1.y + a1.z * b1.z + a1.w * b1.w;
  #pragma unroll
  for (int off = 16; off >= 1; off >>= 1) acc += __shfl_xor(acc, off, 32);

  if (lane == 0) negsim[g] = acc;
}

// ---------------------------------------------------------------------------
// Kernel 3: masked sim-GEMM + hinge-loss epilogue.
// Block = 256 threads (8 waves). Block tile = 16 rows x 128 cols of sim.
// Wave w owns the 16x16 subtile at columns [colBase + 16w, +16).
// A panel (16 x CHUNK) and B panel (128 x CHUNK) staged in LDS per K-chunk.
// ---------------------------------------------------------------------------
__global__ __launch_bounds__(256) void csl_gemm_loss(const float* __restrict__ e,
                                                     const int* __restrict__ ids,
                                                     const float* __restrict__ negsim,
                                                     float* __restrict__ accum) {
  __shared__ float sA[16  * LDP];    //  ~4.3 KB
  __shared__ float sB[128 * LDP];    // ~34.8 KB

  const int t       = threadIdx.x;
  const int wave    = t >> 5;
  const int lane    = t & 31;
  const int rowBase = blockIdx.y * 16;     // 256 tile rows
  const int colBase = blockIdx.x * 128;    // 32 col blocks

  v8f c = {};                              // 16x16 f32 accumulator (8 VGPRs)

  // WMMA fragment addressing (ISA 7.12.2, 32-bit layouts):
  //  A 16x4 : lane L -> row m = L&15, K pair base = (L>>4)*2, comps = K, K+1
  //  B 4x16 : lane L -> col n = L&15, comps hold K = (L>>4)*2 + {0,1}
  //  C 16x16: lane L -> col n = L&15, comp r -> row m = r + (L>>4)*8
  const int am = lane & 15;
  const int kh = (lane >> 4) * 2;
  const int bc = wave * 16 + (lane & 15);  // column (within block tile) this lane reads B for

  for (int kb = 0; kb < D_SZ; kb += CHUNK) {
    // Stage A panel: 16 rows x 64 floats = 256 float4 (1 per thread).
    {
      const int r  = t >> 4;
      const int c4 = (t & 15) * 4;
      float4 v = *(const float4*)(e + (size_t)(rowBase + r) * D_SZ + kb + c4);
      *(float4*)(&sA[r * LDP + c4]) = v;
    }
    // Stage B panel: 128 rows x 64 floats = 2048 float4 (8 per thread).
    #pragma unroll
    for (int p = 0; p < 8; ++p) {
      const int idx = t + p * 256;
      const int r   = idx >> 4;
      const int c4  = (idx & 15) * 4;
      float4 v = *(const float4*)(e + (size_t)(colBase + r) * D_SZ + kb + c4);
      *(float4*)(&sB[r * LDP + c4]) = v;
    }
    __syncthreads();

    #pragma unroll
    for (int kk = 0; kk < CHUNK; kk += 4) {
      const int k0 = kk + kh;
      v2f a = *(const v2f*)(&sA[am * LDP + k0]);
      v2f b = *(const v2f*)(&sB[bc * LDP + k0]);
      // D = A(16x4) * B(4x16) + C, full f32
      c = __builtin_amdgcn_wmma_f32_16x16x4_f32(false, a, false, b,
                                                (short)0, c, false, false);
    }
    __syncthreads();
  }

  // Epilogue: hinge loss over positive (same-speaker, non-diagonal) pairs.
  const int n   = lane & 15;
  const int j   = colBase + wave * 16 + n;
  const int idj = ids[j];
  const int mB  = (lane >> 4) * 8;

  float tot = 0.0f, cnt = 0.0f;
  #pragma unroll
  for (int r = 0; r < 8; ++r) {
    const int i = rowBase + mB + r;
    if (ids[i] == idj && i != j) {
      const float sim = c[r];
      float s = 0.0f;
      #pragma unroll
      for (int k = 0; k < K_NEG; ++k)
        s += fmaxf(0.0f, MARGIN - sim + negsim[i * K_NEG + k]);
      tot += s * (1.0f / (float)K_NEG);
      cnt += 1.0f;
    }
  }

  #pragma unroll
  for (int off = 16; off >= 1; off >>= 1) {
    tot += __shfl_xor(tot, off, 32);
    cnt += __shfl_xor(cnt, off, 32);
  }
  if (lane == 0 && cnt > 0.0f) {
    atomicAdd(&accum[0], tot);
    atomicAdd(&accum[1], cnt);
  }
}

// ---------------------------------------------------------------------------
// Kernel 4: loss = total / max(num_pairs, 1)  (0 if no pairs).
// ---------------------------------------------------------------------------
__global__ void csl_finalize(const float* __restrict__ accum, float* __restrict__ out) {
  const float tot = accum[0];
  const float cnt = accum[1];
  out[0] = (cnt > 0.0f) ? (tot / fmaxf(cnt, 1.0f)) : 0.0f;
}

// ---------------------------------------------------------------------------
// Launch
// ---------------------------------------------------------------------------
extern "C" void kernel_launch(void* const* d_in, const int* in_sizes, int n_in,
                              void* d_out, int out_size, void* d_ws, size_t ws_size,
                              hipStream_t stream) {
  const float* embs   = (const float*)d_in[0];  // (B, D) f32
  const int*   ids    = (const int*)  d_in[1];  // (B,)
  const int*   negidx = (const int*)  d_in[2];  // (B, K)

  float* ws     = (float*)d_ws;
  float* e      = ws;                                  // B*D floats (4 MB)
  float* negsim = ws + (size_t)B_SZ * D_SZ;            // B*K floats
  float* accum  = negsim + (size_t)B_SZ * K_NEG;       // {total, count}
  float* out    = (float*)d_out;

  csl_normalize<<<B_SZ / 8, 256, 0, stream>>>(embs, e, accum);
  csl_negsim<<<(B_SZ * K_NEG) / 8, 256, 0, stream>>>(e, negidx, negsim);

  dim3 grid(B_SZ / 128, B_SZ / 16);   // (32, 256) blocks
  csl_gemm_loss<<<grid, 256, 0, stream>>>(e, ids, negsim, accum);

  csl_finalize<<<1, 1, 0, stream>>>(accum, out);
}